// ALPHALSTM_37692632990006
// MI455X (gfx1250) — compile-verified
//
#include <hip/hip_runtime.h>
#include <hip/hip_bf16.h>

typedef __attribute__((ext_vector_type(2))) float v2f;
typedef __attribute__((ext_vector_type(4))) float v4f;
typedef __attribute__((ext_vector_type(8))) float v8f;
typedef __attribute__((ext_vector_type(4))) unsigned int u32x4;
typedef __attribute__((ext_vector_type(4))) int i32x4;
typedef __attribute__((ext_vector_type(8))) int i32x8;

#if defined(__gfx1250__) && __has_builtin(__builtin_amdgcn_tensor_load_to_lds)
#define HAVE_TDM 1
#endif

// ---------------------------------------------------------------------------
// CDNA5 f32 WMMA: D(16x16) = A(16x4) * B(4x16) + C   (wave32, K=4 per issue)
// ---------------------------------------------------------------------------
__device__ __forceinline__ v8f wmma_f32(v2f a, v2f b, v8f c) {
  return __builtin_amdgcn_wmma_f32_16x16x4_f32(false, a, false, b, (short)0, c,
                                               false, false);
}

__device__ __forceinline__ float fast_tanh(float x) {
  float y;
  asm("v_tanh_f32 %0, %1" : "=v"(y) : "v"(x));
  return y;
}
__device__ __forceinline__ float fast_sigmoid(float x) {
  return __builtin_fmaf(0.5f, fast_tanh(0.5f * x), 0.5f);
}

// A-frag (rows = M) or B-frag (rows = N, weights stored torch-style [out][in]):
// lane&15 selects row; lane>>4 selects K pair {k,k+1} vs {k+2,k+3}.
__device__ __forceinline__ v2f ldfrag(const float* p0, int stride, int row_base,
                                      int k_base, int lane) {
  int r = row_base + (lane & 15);
  int k = k_base + ((lane >> 4) << 1);
  return *(const v2f*)(p0 + (size_t)r * stride + k);
}

#ifdef HAVE_TDM
// TDM: DMA `nelems` contiguous f32 from global `src` into LDS at `lds_ptr`.
// D# built per CDNA5 ISA 8.3/8.4 (1-D tile, data_size=4B, no pad/iterate).
__device__ __forceinline__ void tdm_copy_f32(void* lds_ptr, const void* src,
                                             unsigned nelems) {
  unsigned lds_off = (unsigned)(uintptr_t)lds_ptr;  // flat-LDS low 32 = offset
  unsigned long long ga = (unsigned long long)(uintptr_t)src;
  u32x4 g0;
  g0.x = 1u;                                     // count=1, user descriptor
  g0.y = lds_off;                                // lds_addr (bytes)
  g0.z = (unsigned)ga;                           // global_addr[31:0]
  g0.w = (unsigned)((ga >> 32) & 0x1FFFFFFu) | (2u << 30);  // addr hi | type=2
  i32x8 g1;
  g1[0] = 0x00020000;                            // data_size=2 (4 bytes)
  g1[1] = (int)((nelems & 0xFFFFu) << 16);       // tensor_dim0[15:0]
  g1[2] = (int)((nelems >> 16) & 0xFFFFu) | (1 << 16);  // dim0 hi | dim1=1
  g1[3] = (int)((nelems & 0xFFFFu) << 16);       // tile_dim0 (<=65535)
  g1[4] = 0;                                     // tile_dim1/2 unused
  g1[5] = (int)nelems;                           // tensor_dim0_stride lo
  g1[6] = 0;
  g1[7] = 0;
  i32x4 gz = {0, 0, 0, 0};
#if __clang_major__ >= 23
  i32x8 gz8 = {0, 0, 0, 0, 0, 0, 0, 0};
  __builtin_amdgcn_tensor_load_to_lds(g0, g1, gz, gz, gz8, 0);
#else
  __builtin_amdgcn_tensor_load_to_lds(g0, g1, gz, gz, 0);
#endif
  __builtin_amdgcn_s_wait_tensorcnt(0);
}
#endif

#define SEQ   256
#define BATCH 4096
#define HID   20
#define G4H   80   // 4*HID
#define GSTR  20   // column-major gate scratch stride (bank skew)

// ---------------------------------------------------------------------------
__global__ void zero_kernel(float* __restrict__ p, int n) {
  int i = blockIdx.x * 256 + threadIdx.x;
  if (i < n) p[i] = 0.0f;
}

// ---------------------------------------------------------------------------
// emb MLP: [1M,64] -> relu(50) -> relu(50) -> 20, writes transposed [t][b][20]
// ---------------------------------------------------------------------------
__global__ __launch_bounds__(256) void emb_kernel(
    const float* __restrict__ x,
    const float* __restrict__ W1, const float* __restrict__ b1,
    const float* __restrict__ W2, const float* __restrict__ b2,
    const float* __restrict__ W3, const float* __restrict__ b3,
    float* __restrict__ yout) {
  __shared__ __align__(16) float w1[64 * 64];   // [n<50 pad 64][k=64]
  __shared__ __align__(16) float w2[64 * 52];   // [n<50 pad 64][k<50 pad 52]
  __shared__ __align__(16) float w3[32 * 52];   // [n<20 pad 32][k<50 pad 52]
  __shared__ __align__(16) float bb1[64], bb2[64], bb3[32];
  __shared__ __align__(16) float xs[128 * 64];
  __shared__ __align__(16) float h1[128 * 56];
  __shared__ __align__(16) float h2[128 * 56];

  const int tid  = threadIdx.x;
  const int lane = tid & 31;
  const int wv   = tid >> 5;
  const long tok0 = (long)blockIdx.x * 128;

  for (int i = tid; i < 64 * 64; i += 256) {
    int n = i >> 6, k = i & 63;
    w1[i] = (n < 50) ? W1[n * 64 + k] : 0.0f;
  }
  for (int i = tid; i < 64 * 52; i += 256) {
    int n = i / 52, k = i % 52;
    w2[i] = (n < 50 && k < 50) ? W2[n * 50 + k] : 0.0f;
  }
  for (int i = tid; i < 32 * 52; i += 256) {
    int n = i / 52, k = i % 52;
    w3[i] = (n < 20 && k < 50) ? W3[n * 50 + k] : 0.0f;
  }
  if (tid < 64)       bb1[tid] = (tid < 50) ? b1[tid] : 0.0f;
  else if (tid < 128) { int q = tid - 64;  bb2[q] = (q < 50) ? b2[q] : 0.0f; }
  else if (tid < 160) { int q = tid - 128; bb3[q] = (q < 20) ? b3[q] : 0.0f; }

#ifdef HAVE_TDM
  // Tensor Data Mover: one wave DMAs the whole 128x64 f32 tile into LDS.
  if (wv == 0) tdm_copy_f32(xs, x + tok0 * 64, 128 * 64);
#else
  for (int i = tid; i < 128 * 64; i += 256) xs[i] = x[tok0 * 64 + i];
#endif
  __syncthreads();

  const int mb = wv * 16;  // wave-private 16 token rows; no more barriers

  // stage 1: 64 -> 50 (padded N=64), K=64
  for (int nt = 0; nt < 4; ++nt) {
    float bias = bb1[nt * 16 + (lane & 15)];
    v8f acc;
    #pragma unroll
    for (int j = 0; j < 8; ++j) acc[j] = bias;
    for (int kk = 0; kk < 16; ++kk) {
      v2f a = ldfrag(xs, 64, mb, kk * 4, lane);
      v2f b = ldfrag(w1, 64, nt * 16, kk * 4, lane);
      acc = wmma_f32(a, b, acc);
    }
    #pragma unroll
    for (int j = 0; j < 8; ++j) {
      int m = mb + j + ((lane >> 4) << 3);
      int n = nt * 16 + (lane & 15);
      if (n < 52) h1[m * 56 + n] = fmaxf(acc[j], 0.0f);
    }
  }
  // stage 2: 50 -> 50, K padded 52
  for (int nt = 0; nt < 4; ++nt) {
    float bias = bb2[nt * 16 + (lane & 15)];
    v8f acc;
    #pragma unroll
    for (int j = 0; j < 8; ++j) acc[j] = bias;
    for (int kk = 0; kk < 13; ++kk) {
      v2f a = ldfrag(h1, 56, mb, kk * 4, lane);
      v2f b = ldfrag(w2, 52, nt * 16, kk * 4, lane);
      acc = wmma_f32(a, b, acc);
    }
    #pragma unroll
    for (int j = 0; j < 8; ++j) {
      int m = mb + j + ((lane >> 4) << 3);
      int n = nt * 16 + (lane & 15);
      if (n < 52) h2[m * 56 + n] = fmaxf(acc[j], 0.0f);
    }
  }
  // stage 3: 50 -> 20, write transposed into y[t=n_idx][b][20]
  for (int nt = 0; nt < 2; ++nt) {
    float bias = bb3[nt * 16 + (lane & 15)];
    v8f acc;
    #pragma unroll
    for (int j = 0; j < 8; ++j) acc[j] = bias;
    for (int kk = 0; kk < 13; ++kk) {
      v2f a = ldfrag(h2, 56, mb, kk * 4, lane);
      v2f b = ldfrag(w3, 52, nt * 16, kk * 4, lane);
      acc = wmma_f32(a, b, acc);
    }
    #pragma unroll
    for (int j = 0; j < 8; ++j) {
      int m = mb + j + ((lane >> 4) << 3);
      int n = nt * 16 + (lane & 15);
      if (n < HID) {
        long tok = tok0 + m;
        long bi = tok >> 8;        // token / SEQ
        long ni = tok & 255;       // token % SEQ
        yout[(ni * BATCH + bi) * HID + n] = acc[j];
      }
    }
  }
}

// ---------------------------------------------------------------------------
// 8 stacked LSTM layers, fully fused. Each wave owns 16 batch rows and walks
// (layer, t) with weight B-fragments AND bias accumulators resident in VGPRs.
// ---------------------------------------------------------------------------
struct LstmWeights { const float* p[32]; };  // 8 x {Wih, Whh, bih, bhh}

__global__ __launch_bounds__(256) void lstm_kernel(LstmWeights Wt,
                                                   float* __restrict__ ping,
                                                   float* __restrict__ pong) {
  __shared__ __align__(16) float g_s[8][G4H * GSTR];  // column-major [n][m]
  __shared__ __align__(16) float h_s[8][16 * HID];
  __shared__ __align__(16) float c_s[8][16 * HID];

  const int tid  = threadIdx.x;
  const int lane = tid & 31;
  const int wv   = tid >> 5;
  float* G  = g_s[wv];
  float* Hh = h_s[wv];
  float* Cc = c_s[wv];
  const int row0 = blockIdx.x * 128 + wv * 16;
  const int mi   = lane & 15;
  const int koff = (lane >> 4) << 1;
  const int joff = (lane >> 4) << 3;

  for (int layer = 0; layer < 8; ++layer) {
    const float* Wih = Wt.p[layer * 4 + 0];
    const float* Whh = Wt.p[layer * 4 + 1];
    const float* bih = Wt.p[layer * 4 + 2];
    const float* bhh = Wt.p[layer * 4 + 3];

    v2f fih[5][5], fhh[5][5];
    v8f cinit[5];                       // bias broadcast, persistent C operand
    #pragma unroll
    for (int nt = 0; nt < 5; ++nt) {
      float s = bih[nt * 16 + mi] + bhh[nt * 16 + mi];
      #pragma unroll
      for (int j = 0; j < 8; ++j) cinit[nt][j] = s;
      #pragma unroll
      for (int kk = 0; kk < 5; ++kk) {
        fih[nt][kk] = ldfrag(Wih, HID, nt * 16, kk * 4, lane);
        fhh[nt][kk] = ldfrag(Whh, HID, nt * 16, kk * 4, lane);
      }
    }
    for (int q = lane; q < 16 * HID; q += 32) { Hh[q] = 0.0f; Cc[q] = 0.0f; }

    const float* yin = (layer & 1) ? pong : ping;
    float*      yout = (layer & 1) ? ping : pong;

    for (int t = 0; t < SEQ + 1; ++t) {
      const float* xr = yin + ((size_t)t * BATCH + row0 + mi) * HID + koff;
      if (t < SEQ)  // hide next step's global x fetch behind this step
        __builtin_prefetch(xr + (size_t)BATCH * HID, 0, 3);

      v8f acc[5];
      // kk = 0 chains directly off the persistent bias accumulators: no movs
      {
        v2f ax = *(const v2f*)(xr);
        v2f ah = *(const v2f*)(Hh + mi * HID + koff);
        #pragma unroll
        for (int nt = 0; nt < 5; ++nt)
          acc[nt] = wmma_f32(ax, fih[nt][0], cinit[nt]);
        #pragma unroll
        for (int nt = 0; nt < 5; ++nt)
          acc[nt] = wmma_f32(ah, fhh[nt][0], acc[nt]);
      }
      #pragma unroll
      for (int kk = 1; kk < 5; ++kk) {
        v2f ax = *(const v2f*)(xr + kk * 4);
        v2f ah = *(const v2f*)(Hh + mi * HID + kk * 4 + koff);
        #pragma unroll
        for (int nt = 0; nt < 5; ++nt) {
          acc[nt] = wmma_f32(ax, fih[nt][kk], acc[nt]);
          acc[nt] = wmma_f32(ah, fhh[nt][kk], acc[nt]);
        }
      }
      // spill gates column-major: 8 m-contiguous elems -> two ds_store_b128
      #pragma unroll
      for (int nt = 0; nt < 5; ++nt) {
        float* gp = G + (nt * 16 + mi) * GSTR + joff;
        *(v4f*)(gp)     = __builtin_shufflevector(acc[nt], acc[nt], 0, 1, 2, 3);
        *(v4f*)(gp + 4) = __builtin_shufflevector(acc[nt], acc[nt], 4, 5, 6, 7);
      }
      // gate nonlinearities: i,f,g,o at columns q, 20+q, 40+q, 60+q
      #pragma unroll
      for (int e = 0; e < 10; ++e) {
        int idx = e * 32 + lane;       // 0..319 covers 16 rows x 20 cols
        int mm = idx / HID;
        int q  = idx - mm * HID;
        float gi = G[(q)      * GSTR + mm];
        float gf = G[(20 + q) * GSTR + mm];
        float gg = G[(40 + q) * GSTR + mm];
        float go = G[(60 + q) * GSTR + mm];
        float cn = fast_sigmoid(gf) * Cc[mm * HID + q] +
                   fast_sigmoid(gi) * fast_tanh(gg);
        Cc[mm * HID + q] = cn;
        float hn = fast_sigmoid(go) * fast_tanh(cn);
        Hh[mm * HID + q] = hn;
        yout[((size_t)t * BATCH + row0 + mm) * HID + q] = hn;
      }
    }
  }
}

// ---------------------------------------------------------------------------
// proj MLP on last timestep: [4096,20] -> relu(50) -> relu(50) -> 29
// ---------------------------------------------------------------------------
__global__ __launch_bounds__(256) void proj_kernel(
    const float* __restrict__ last,  // [4096][20], = y_final[t=256]
    const float* __restrict__ W1, const float* __restrict__ b1,
    const float* __restrict__ W2, const float* __restrict__ b2,
    const float* __restrict__ W3, const float* __restrict__ b3,
    float* __restrict__ out) {
  __shared__ __align__(16) float w1[64 * 20];   // [n<50 pad 64][k=20]
  __shared__ __align__(16) float w2[64 * 52];
  __shared__ __align__(16) float w3[32 * 52];
  __shared__ __align__(16) float bb1[64], bb2[64], bb3[32];
  __shared__ __align__(16) float xs[128 * 20];
  __shared__ __align__(16) float h1[128 * 56];
  __shared__ __align__(16) float h2[128 * 56];

  const int tid  = threadIdx.x;
  const int lane = tid & 31;
  const int wv   = tid >> 5;
  const long tok0 = (long)blockIdx.x * 128;

  for (int i = tid; i < 64 * 20; i += 256) {
    int n = i / 20, k = i % 20;
    w1[i] = (n < 50) ? W1[n * 20 + k] : 0.0f;
  }
  for (int i = tid; i < 64 * 52; i += 256) {
    int n = i / 52, k = i % 52;
    w2[i] = (n < 50 && k < 50) ? W2[n * 50 + k] : 0.0f;
  }
  for (int i = tid; i < 32 * 52; i += 256) {
    int n = i / 52, k = i % 52;
    w3[i] = (n < 29 && k < 50) ? W3[n * 50 + k] : 0.0f;
  }
  if (tid < 64)       bb1[tid] = (tid < 50) ? b1[tid] : 0.0f;
  else if (tid < 128) { int q = tid - 64;  bb2[q] = (q < 50) ? b2[q] : 0.0f; }
  else if (tid < 160) { int q = tid - 128; bb3[q] = (q < 29) ? b3[q] : 0.0f; }
#ifdef HAVE_TDM
  if (wv == 0) tdm_copy_f32(xs, last + tok0 * 20, 128 * 20);
#else
  for (int i = tid; i < 128 * 20; i += 256) xs[i] = last[tok0 * 20 + i];
#endif
  __syncthreads();

  const int mb = wv * 16;

  for (int nt = 0; nt < 4; ++nt) {
    float bias = bb1[nt * 16 + (lane & 15)];
    v8f acc;
    #pragma unroll
    for (int j = 0; j < 8; ++j) acc[j] = bias;
    for (int kk = 0; kk < 5; ++kk) {
      v2f a = ldfrag(xs, 20, mb, kk * 4, lane);
      v2f b = ldfrag(w1, 20, nt * 16, kk * 4, lane);
      acc = wmma_f32(a, b, acc);
    }
    #pragma unroll
    for (int j = 0; j < 8; ++j) {
      int m = mb + j + ((lane >> 4) << 3);
      int n = nt * 16 + (lane & 15);
      if (n < 52) h1[m * 56 + n] = fmaxf(acc[j], 0.0f);
    }
  }
  for (int nt = 0; nt < 4; ++nt) {
    float bias = bb2[nt * 16 + (lane & 15)];
    v8f acc;
    #pragma unroll
    for (int j = 0; j < 8; ++j) acc[j] = bias;
    for (int kk = 0; kk < 13; ++kk) {
      v2f a = ldfrag(h1, 56, mb, kk * 4, lane);
      v2f b = ldfrag(w2, 52, nt * 16, kk * 4, lane);
      acc = wmma_f32(a, b, acc);
    }
    #pragma unroll
    for (int j = 0; j < 8; ++j) {
      int m = mb + j + ((lane >> 4) << 3);
      int n = nt * 16 + (lane & 15);
      if (n < 52) h2[m * 56 + n] = fmaxf(acc[j], 0.0f);
    }
  }
  for (int nt = 0; nt < 2; ++nt) {
    float bias = bb3[nt * 16 + (lane & 15)];
    v8f acc;
    #pragma unroll
    for (int j = 0; j < 8; ++j) acc[j] = bias;
    for (int kk = 0; kk < 13; ++kk) {
      v2f a = ldfrag(h2, 56, mb, kk * 4, lane);
      v2f b = ldfrag(w3, 52, nt * 16, kk * 4, lane);
      acc = wmma_f32(a, b, acc);
    }
    #pragma unroll
    for (int j = 0; j < 8; ++j) {
      int m = mb + j + ((lane >> 4) << 3);
      int n = nt * 16 + (lane & 15);
      if (n < 29) out[(tok0 + m) * 29 + n] = acc[j];
    }
  }
}

// ---------------------------------------------------------------------------
extern "C" void kernel_launch(void* const* d_in, const int* in_sizes, int n_in,
                              void* d_out, int out_size, void* d_ws,
                              size_t ws_size, hipStream_t stream) {
  (void)in_sizes; (void)n_in; (void)out_size; (void)ws_size;

  const float* x   = (const float*)d_in[0];
  const float* eW1 = (const float*)d_in[1];
  const float* eb1 = (const float*)d_in[2];
  const float* eW2 = (const float*)d_in[3];
  const float* eb2 = (const float*)d_in[4];
  const float* eW3 = (const float*)d_in[5];
  const float* eb3 = (const float*)d_in[6];
  // d_in[7..38]: 8 x (Wih, Whh, bih, bhh)
  const float* pW1 = (const float*)d_in[39];
  const float* pb1 = (const float*)d_in[40];
  const float* pW2 = (const float*)d_in[41];
  const float* pb2 = (const float*)d_in[42];
  const float* pW3 = (const float*)d_in[43];
  const float* pb3 = (const float*)d_in[44];

  const size_t buf_elems = (size_t)(SEQ + 1) * BATCH * HID;  // 257*4096*20
  float* ping = (float*)d_ws;
  float* pong = ping + buf_elems;

  // zero pad timestep t=256 of the first activation buffer
  zero_kernel<<<(BATCH * HID + 255) / 256, 256, 0, stream>>>(
      ping + (size_t)SEQ * BATCH * HID, BATCH * HID);

  // embedding MLP -> ping, already transposed to [t][b][20]
  emb_kernel<<<(BATCH * SEQ) / 128, 256, 0, stream>>>(x, eW1, eb1, eW2, eb2,
                                                      eW3, eb3, ping);

  // fused 8-layer LSTM stack (ping <-> pong; layer 7 output lands in ping)
  LstmWeights Wt;
  for (int i = 0; i < 32; ++i) Wt.p[i] = (const float*)d_in[7 + i];
  lstm_kernel<<<BATCH / 128, 256, 0, stream>>>(Wt, ping, pong);

  // projection MLP on last timestep
  proj_kernel<<<BATCH / 128, 256, 0, stream>>>(
      ping + (size_t)SEQ * BATCH * HID, pW1, pb1, pW2, pb2, pW3, pb3,
      (float*)d_out);
}